// DownLayer_3650722202090
// MI455X (gfx1250) — compile-verified
//
#include <hip/hip_runtime.h>

// Problem constants (from reference: x (1,2048,11008) f32, weight (4096,11008) f32)
#define M_DIM 2048
#define N_DIM 4096
#define K_DIM 11008
#define K_TOK 2201   // int(11008*0.2)
#define N_CORE 4403  // int(11008*0.4)

typedef __attribute__((ext_vector_type(2))) float v2f;
typedef __attribute__((ext_vector_type(8))) float v8f;

// ---------------------------------------------------------------------------
// GEMM: out[m][n] = sum_k x[m][k] * w[n][k]   (fp32, V_WMMA_F32_16X16X4_F32)
// Block tile 128x128, BK=32, 256 threads = 8 waves in a 2(M) x 4(N) grid.
// Each wave owns a 64x32 sub-tile = 4x2 WMMA 16x16 accumulators.
// Double-buffered dense LDS (XOR bank swizzle, 64KB total), staged with
// GLOBAL_LOAD_ASYNC_TO_LDS_B128 and synchronized via ASYNCcnt.
// ---------------------------------------------------------------------------
#define BM 128
#define BN 128
#define BK 32

// XOR swizzle: logical (row, col) -> physical float index within a tile.
// Moves whole 4-float blocks, so B128 stores stay 16B aligned and float2
// fragment reads stay 8B aligned; conflict-free for 16-row fragment reads.
__device__ __forceinline__ int swz(int row, int blk) {  // blk = col>>2
  return row * BK + (((blk ^ ((row >> 1) & 7)) << 2));
}

__global__ __launch_bounds__(256)
void gemm_wmma_f32(const float* __restrict__ X, const float* __restrict__ W,
                   float* __restrict__ out) {
  __shared__ float As[2][BM * BK];  // 2 x 16384 B
  __shared__ float Bs[2][BN * BK];  // 2 x 16384 B  (total = 64 KB)

  const int tid   = threadIdx.x;
  const int lane  = tid & 31;
  const int wid   = tid >> 5;     // 0..7
  const int waveM = wid & 1;      // 2 waves along M
  const int waveN = wid >> 1;     // 4 waves along N
  const int blockM = blockIdx.y * BM;
  const int blockN = blockIdx.x * BN;

  const int lr   = lane & 15;        // row within 16-wide fragment
  const int koff = (lane >> 4) << 1; // K offset 0 or 2 (A/B frag layout)
  const int moff = (lane >> 4) << 3; // M offset 0 or 8 (C/D layout)
  // Swizzle key is identical for all fragments this lane reads:
  // ((m>>1)&7)<<2 with m = 16*c + lr  ==  ((lr>>1)&7)<<2.
  const int key4 = ((lr >> 1) & 7) << 2;

  // Async DMA of one BMxBK (and BNxBK) tile into LDS buffer `buf`.
  // 1024 B128 slots per tile / 256 threads = 4 async ops each, per matrix.
  auto stage = [&](int buf, int k0) {
#pragma unroll
    for (int i = 0; i < 4; ++i) {
      const int f   = tid + i * 256;  // 0..1023
      const int row = f >> 3;         // 0..127 (8 blocks per 32-float row)
      const int blk = f & 7;          // 4-float block index
      const int p   = swz(row, blk);
      const float* ga = X + (size_t)(blockM + row) * K_DIM + k0 + (blk << 2);
      const unsigned la = (unsigned)(uintptr_t)&As[buf][p];
      asm volatile("global_load_async_to_lds_b128 %0, %1, off"
                   :: "v"(la), "v"(ga) : "memory");
      const float* gb = W + (size_t)(blockN + row) * K_DIM + k0 + (blk << 2);
      const unsigned lb = (unsigned)(uintptr_t)&Bs[buf][p];
      asm volatile("global_load_async_to_lds_b128 %0, %1, off"
                   :: "v"(lb), "v"(gb) : "memory");
    }
  };

  v8f acc[4][2];
#pragma unroll
  for (int mi = 0; mi < 4; ++mi)
#pragma unroll
    for (int ni = 0; ni < 2; ++ni)
#pragma unroll
      for (int v = 0; v < 8; ++v) acc[mi][ni][v] = 0.0f;

  // Per-fragment base float indices (lane-varying, loop-invariant).
  int baseA[4], baseB[2];
#pragma unroll
  for (int mi = 0; mi < 4; ++mi)
    baseA[mi] = (waveM * 64 + mi * 16 + lr) * BK + koff;
#pragma unroll
  for (int ni = 0; ni < 2; ++ni)
    baseB[ni] = (waveN * 32 + ni * 16 + lr) * BK + koff;

  // Prologue: fill buffer 0.
  stage(0, 0);
  asm volatile("s_wait_asynccnt 0x0" ::: "memory");
  __syncthreads();

  int cur = 0;
  for (int k0 = 0; k0 < K_DIM; k0 += BK) {
    // Prefetch next tile into the other buffer while computing this one.
    if (k0 + BK < K_DIM) stage(cur ^ 1, k0 + BK);

    const float* Ab = &As[cur][0];
    const float* Bb = &Bs[cur][0];
#pragma unroll
    for (int ks = 0; ks < BK; ks += 4) {
      const int xk = ks ^ key4;  // == ((ks>>2) ^ key)<<2, shared by all frags
      v2f a[4], b[2];
#pragma unroll
      for (int mi = 0; mi < 4; ++mi) a[mi] = *(const v2f*)(&Ab[baseA[mi] + xk]);
#pragma unroll
      for (int ni = 0; ni < 2; ++ni) b[ni] = *(const v2f*)(&Bb[baseB[ni] + xk]);
#pragma unroll
      for (int mi = 0; mi < 4; ++mi)
#pragma unroll
        for (int ni = 0; ni < 2; ++ni)
          acc[mi][ni] = __builtin_amdgcn_wmma_f32_16x16x4_f32(
              false, a[mi], false, b[ni], (short)0, acc[mi][ni], false, false);
    }

    // Our async prefetch must have landed before anyone reads the new buffer.
    asm volatile("s_wait_asynccnt 0x0" ::: "memory");
    __syncthreads();
    cur ^= 1;
  }

  // Epilogue: C/D layout — VGPR v holds M = v (+8 for lanes 16..31), N = lane&15
#pragma unroll
  for (int mi = 0; mi < 4; ++mi)
#pragma unroll
    for (int ni = 0; ni < 2; ++ni) {
      const int nIdx  = blockN + waveN * 32 + ni * 16 + lr;
      const int mBase = blockM + waveM * 64 + mi * 16 + moff;
#pragma unroll
      for (int v = 0; v < 8; ++v)
        out[(size_t)(mBase + v) * N_DIM + nIdx] = acc[mi][ni][v];
    }
}

// ---------------------------------------------------------------------------
// Selection pipeline
// ---------------------------------------------------------------------------
__device__ __forceinline__ unsigned fkey(float f) {
  unsigned u = __float_as_uint(f);
  return (u & 0x80000000u) ? ~u : (u | 0x80000000u);  // monotonic order key
}

// Per-token exact k-th largest via 4-pass radix select on the 32-bit key.
__global__ __launch_bounds__(256)
void topk_threshold(const float* __restrict__ X, unsigned* __restrict__ thr) {
  __shared__ unsigned hist[256];
  __shared__ unsigned pref_s;
  __shared__ int krem_s;
  const int t = blockIdx.x;
  const float* xr = X + (size_t)t * K_DIM;
  unsigned prefix = 0;
  int krem = K_TOK;
  for (int pass = 3; pass >= 0; --pass) {
    const int shift = pass * 8;
    hist[threadIdx.x] = 0;
    __syncthreads();
    for (int i = threadIdx.x; i < K_DIM; i += 256) {
      const unsigned u = fkey(xr[i]);
      const bool match =
          (pass == 3) || ((u >> (shift + 8)) == (prefix >> (shift + 8)));
      if (match) atomicAdd(&hist[(u >> shift) & 0xFFu], 1u);
    }
    __syncthreads();
    if (threadIdx.x == 0) {
      unsigned cum = 0, d = 0;
      for (int dd = 255; dd >= 0; --dd) {
        cum += hist[dd];
        if ((int)cum >= krem) { d = (unsigned)dd; break; }
      }
      pref_s = prefix | (d << shift);
      krem_s = krem - (int)(cum - hist[d]);
    }
    __syncthreads();
    prefix = pref_s;
    krem = krem_s;
    __syncthreads();
  }
  if (threadIdx.x == 0) thr[t] = prefix;
}

__global__ void zero_counts(int* __restrict__ counts) {
  const int i = blockIdx.x * blockDim.x + threadIdx.x;
  if (i < K_DIM) counts[i] = 0;
}

__global__ __launch_bounds__(256)
void count_core(const float* __restrict__ X, const unsigned* __restrict__ thr,
                int* __restrict__ counts) {
  const int t = blockIdx.x;
  const unsigned T = thr[t];
  const float* xr = X + (size_t)t * K_DIM;
  for (int i = threadIdx.x; i < K_DIM; i += 256)
    if (fkey(xr[i]) >= T) atomicAdd(&counts[i], 1);
}

// Exact top-N_CORE of counts with jax top_k tie-breaking (lower index first):
// total order key = (count << 14) | (K_DIM-1-n); rank = #keys greater.
__global__ __launch_bounds__(256)
void rank_select(const int* __restrict__ counts, int* __restrict__ core_idx) {
  __shared__ int sc[K_DIM];  // 44032 B
  for (int i = threadIdx.x; i < K_DIM; i += 256) sc[i] = counts[i];
  __syncthreads();
  const int n = blockIdx.x * 256 + threadIdx.x;
  if (n >= K_DIM) return;
  const unsigned mykey =
      ((unsigned)sc[n] << 14) | (unsigned)(K_DIM - 1 - n);
  int rank = 0;
  for (int j = 0; j < K_DIM; ++j) {
    const unsigned kj = ((unsigned)sc[j] << 14) | (unsigned)(K_DIM - 1 - j);
    rank += (kj > mykey) ? 1 : 0;
  }
  if (rank < N_CORE) core_idx[rank] = n;
}

__global__ void gather_w(const float* __restrict__ W,
                         const int* __restrict__ core_idx,
                         float* __restrict__ outF) {
  const long long i = (long long)blockIdx.x * blockDim.x + threadIdx.x;
  const long long total = (long long)N_DIM * N_CORE;
  if (i >= total) return;
  const int d = (int)(i / N_CORE);
  const int j = (int)(i - (long long)d * N_CORE);
  outF[i] = W[(size_t)d * K_DIM + core_idx[j]];
}

// ---------------------------------------------------------------------------
extern "C" void kernel_launch(void* const* d_in, const int* in_sizes, int n_in,
                              void* d_out, int out_size, void* d_ws,
                              size_t ws_size, hipStream_t stream) {
  const float* X = (const float*)d_in[0];  // (1,2048,11008) f32
  const float* W = (const float*)d_in[1];  // (4096,11008) f32
  float* out  = (float*)d_out;                       // true_value (2048x4096)
  float* outF = out + (size_t)M_DIM * N_DIM;         // filtered_W (4096x4403)

  unsigned* thr  = (unsigned*)d_ws;                        // 2048 u32
  int* counts    = (int*)((char*)d_ws + (16 << 10));       // 11008 i32
  int* core_idx  = (int*)((char*)d_ws + (80 << 10));       // 4403 i32

  dim3 gGemm(N_DIM / BN, M_DIM / BM);  // 32 x 16
  gemm_wmma_f32<<<gGemm, 256, 0, stream>>>(X, W, out);

  topk_threshold<<<M_DIM, 256, 0, stream>>>(X, thr);
  zero_counts<<<(K_DIM + 255) / 256, 256, 0, stream>>>(counts);
  count_core<<<M_DIM, 256, 0, stream>>>(X, thr, counts);
  rank_select<<<(K_DIM + 255) / 256, 256, 0, stream>>>(counts, core_idx);

  const long long total = (long long)N_DIM * N_CORE;
  gather_w<<<(int)((total + 255) / 256), 256, 0, stream>>>(W, core_idx, outF);
}